// DenseInputEncoder_72610717106356
// MI455X (gfx1250) — compile-verified
//
#include <hip/hip_runtime.h>

typedef __attribute__((ext_vector_type(2))) float v2f;
typedef __attribute__((ext_vector_type(4))) float v4f;
typedef __attribute__((ext_vector_type(8))) float v8f;

// Problem constants (match reference file)
constexpr int BG     = 64;           // graphs
constexpr int NMAX   = 128;          // dense padding per graph
constexpr int HD     = 64;           // hidden dim
constexpr int NT     = 4096;         // total nodes
constexpr int NE     = 65536;        // edges / pairs
constexpr int PLANE  = NMAX * NMAX;          // 16384
constexpr int GSTR   = HD * PLANE;           // 1048576 floats per graph
constexpr int OUT_PAIR = BG * GSTR;          // 67108864 floats, mask follows

// ---------------------------------------------------------------------------
// 1) Zero the entire output (b128 stores; bandwidth-bound part of this op)
//    and the first 128 ints of workspace (counts + starts).
// ---------------------------------------------------------------------------
__global__ void zero_kernel(float* __restrict__ out, int n_f4, int n_total,
                            int* __restrict__ ws) {
  int i = blockIdx.x * blockDim.x + threadIdx.x;
  int stride = gridDim.x * blockDim.x;
  v4f z = {0.f, 0.f, 0.f, 0.f};
  for (int k = i; k < n_f4; k += stride)
    reinterpret_cast<v4f*>(out)[k] = z;
  if (i == 0)
    for (int k = n_f4 * 4; k < n_total; ++k) out[k] = 0.f;
  if (i < 128) ws[i] = 0;
}

// ---------------------------------------------------------------------------
// 2) Per-graph node counts (segment_sum of ones over `batch`).
// ---------------------------------------------------------------------------
__global__ void count_kernel(const int* __restrict__ batch, int* __restrict__ ws) {
  int i = blockIdx.x * blockDim.x + threadIdx.x;
  int stride = gridDim.x * blockDim.x;
  for (int k = i; k < NT; k += stride)
    atomicAdd(&ws[batch[k]], 1);
}

// 3) Tiny exclusive scan: starts[b] = sum(counts[0..b-1]); starts at ws+64.
__global__ void scan_kernel(int* __restrict__ ws) {
  int s = 0;
  for (int g = 0; g < BG; ++g) { ws[64 + g] = s; s += ws[g]; }
}

// ---------------------------------------------------------------------------
// 4) Node encoder: node_x@W_node + loop_x@W_loop via WMMA f32 16x16x4,
//    atomically added onto the diagonal of out[b,h,:,:]; writes mask.
//    One wave handles 16 nodes x 64 H.
// ---------------------------------------------------------------------------
__global__ void node_kernel(const float* __restrict__ node_x,
                            const float* __restrict__ loop_x,
                            const float* __restrict__ Wn,
                            const float* __restrict__ Wl,
                            const int* __restrict__ batch,
                            const int* __restrict__ ws,
                            float* __restrict__ out) {
  const int* starts = ws + 64;
  int lane = threadIdx.x & 31;
  int wave = (blockIdx.x * blockDim.x + threadIdx.x) >> 5;
  int n0 = wave * 16;
  if (n0 >= NT) return;                 // wave-uniform: EXEC stays all-ones
  const bool hi = lane >= 16;
  const int lm = lane & 15;
  const int koff = hi ? 2 : 0;

  // A fragments: 16x4 f32 layout — lanes 0-15: row=lane, K=k0..k0+1;
  //              lanes 16-31: row=lane-16, K=k0+2..k0+3.
  v2f an[8];
  {
    const float* p = node_x + (size_t)(n0 + lm) * 32 + koff;
#pragma unroll
    for (int j = 0; j < 8; ++j) an[j] = *reinterpret_cast<const v2f*>(p + 4 * j);
  }
  v2f al[4];
  {
    const float* p = loop_x + (size_t)(n0 + lm) * 16 + koff;
#pragma unroll
    for (int j = 0; j < 4; ++j) al[j] = *reinterpret_cast<const v2f*>(p + 4 * j);
  }

  // Per-lane scatter targets: VGPR v of the C tile holds node row v (lanes
  // 0-15) or row v+8 (lanes 16-31); lane%16 selects the H column.
  int tb[8], mb[8];
  const int nb = n0 + (hi ? 8 : 0);
#pragma unroll
  for (int v = 0; v < 8; ++v) {
    int n = nb + v;
    int g = batch[n];
    int p = n - starts[g];
    tb[v] = g * GSTR + p * NMAX + p;   // diagonal element, h added later
    mb[v] = g * NMAX + p;
  }

#pragma unroll
  for (int nt = 0; nt < 4; ++nt) {
    const int h = nt * 16 + lm;
    v8f c = {};
#pragma unroll
    for (int j = 0; j < 8; ++j) {      // K=32 node part
      v2f bf;
      bf.x = Wn[(4 * j + koff) * HD + h];
      bf.y = Wn[(4 * j + 1 + koff) * HD + h];
      c = __builtin_amdgcn_wmma_f32_16x16x4_f32(false, an[j], false, bf,
                                                (short)0, c, false, false);
    }
#pragma unroll
    for (int j = 0; j < 4; ++j) {      // K=16 loop part, same accumulator
      v2f bf;
      bf.x = Wl[(4 * j + koff) * HD + h];
      bf.y = Wl[(4 * j + 1 + koff) * HD + h];
      c = __builtin_amdgcn_wmma_f32_16x16x4_f32(false, al[j], false, bf,
                                                (short)0, c, false, false);
    }
#pragma unroll
    for (int v = 0; v < 8; ++v)
      atomicAdd(out + tb[v] + h * PLANE, c[v]);
  }

  if (lm == 0) {                        // lanes 0 and 16 cover rows 0-7 / 8-15
    float* mask = out + OUT_PAIR;
#pragma unroll
    for (int v = 0; v < 8; ++v) mask[mb[v]] = 1.0f;
  }
}

// ---------------------------------------------------------------------------
// 5) Edge/pair encoder + dense-adj scatter-add, fused. One wave handles
//    16 edges x 64 H via K/4 chained WMMA f32 16x16x4 ops per 16-wide H tile.
// ---------------------------------------------------------------------------
template <int K>
__global__ void scatter_kernel(const float* __restrict__ attr,  // [NE,K]
                               const float* __restrict__ W,     // [K,HD]
                               const int* __restrict__ eidx,    // [2,NE]
                               const int* __restrict__ batch,
                               const int* __restrict__ ws,
                               float* __restrict__ out) {
  const int* starts = ws + 64;
  int lane = threadIdx.x & 31;
  int wave = (blockIdx.x * blockDim.x + threadIdx.x) >> 5;
  int e0 = wave * 16;
  if (e0 >= NE) return;                 // wave-uniform
  const bool hi = lane >= 16;
  const int lm = lane & 15;
  const int koff = hi ? 2 : 0;
  constexpr int KS = K / 4;

  v2f a[KS];
  {
    const float* p = attr + (size_t)(e0 + lm) * K + koff;
#pragma unroll
    for (int j = 0; j < KS; ++j) a[j] = *reinterpret_cast<const v2f*>(p + 4 * j);
  }

  // Per-lane targets: reference is out[batch[src], pos[src], pos[dst]] (+= h)
  int tb[8];
  const int eb = e0 + (hi ? 8 : 0);
#pragma unroll
  for (int v = 0; v < 8; ++v) {
    int e = eb + v;
    int s = eidx[e];
    int d = eidx[NE + e];
    int g = batch[s];
    int r = s - starts[g];
    int cc = d - starts[batch[d]];
    tb[v] = g * GSTR + r * NMAX + cc;
  }

#pragma unroll
  for (int nt = 0; nt < 4; ++nt) {
    const int h = nt * 16 + lm;
    v8f c = {};
#pragma unroll
    for (int j = 0; j < KS; ++j) {
      v2f bf;
      bf.x = W[(4 * j + koff) * HD + h];
      bf.y = W[(4 * j + 1 + koff) * HD + h];
      c = __builtin_amdgcn_wmma_f32_16x16x4_f32(false, a[j], false, bf,
                                                (short)0, c, false, false);
    }
#pragma unroll
    for (int v = 0; v < 8; ++v)
      atomicAdd(out + tb[v] + h * PLANE, c[v]);
  }
}

// ---------------------------------------------------------------------------
extern "C" void kernel_launch(void* const* d_in, const int* in_sizes, int n_in,
                              void* d_out, int out_size, void* d_ws, size_t ws_size,
                              hipStream_t stream) {
  const float* node_x    = (const float*)d_in[0];
  const float* loop_x    = (const float*)d_in[1];
  const float* edge_attr = (const float*)d_in[2];
  const float* pair_x    = (const float*)d_in[3];
  const float* W_node    = (const float*)d_in[4];
  const float* W_loop    = (const float*)d_in[5];
  const float* W_edge    = (const float*)d_in[6];
  const float* W_pair    = (const float*)d_in[7];
  const int*   batch     = (const int*)d_in[8];
  const int*   eidx      = (const int*)d_in[9];
  const int*   pidx      = (const int*)d_in[10];
  float* out = (float*)d_out;
  int*   ws  = (int*)d_ws;

  int n_f4 = out_size / 4;
  zero_kernel<<<4096, 256, 0, stream>>>(out, n_f4, out_size, ws);
  count_kernel<<<16, 256, 0, stream>>>(batch, ws);
  scan_kernel<<<1, 1, 0, stream>>>(ws);
  // NT/16 = 256 waves, 8 waves (256 threads) per block -> 32 blocks
  node_kernel<<<32, 256, 0, stream>>>(node_x, loop_x, W_node, W_loop, batch, ws, out);
  // NE/16 = 4096 waves -> 512 blocks
  scatter_kernel<32><<<512, 256, 0, stream>>>(edge_attr, W_edge, eidx, batch, ws, out);
  scatter_kernel<16><<<512, 256, 0, stream>>>(pair_x,    W_pair, pidx, batch, ws, out);
}